// CifPredictor_30631706755486
// MI455X (gfx1250) — compile-verified
//
#include <hip/hip_runtime.h>
#include <math.h>

#define B_   16
#define T_   1024
#define D_   512
#define T1_  1025
#define TAIL_ 0.45f
#define SHSTRIDE 516   // floats; 516*4=2064 B (16B multiple), 516%64==4 -> conflict-free

typedef __attribute__((ext_vector_type(2))) float v2f;
typedef __attribute__((ext_vector_type(8))) float v8f;
typedef __attribute__((ext_vector_type(4))) int   v4i;

#if defined(__AMDGCN__) && __has_builtin(__builtin_amdgcn_global_load_async_to_lds_b128)
#define HAVE_ASYNC_LDS 1
#endif

// ---------------------------------------------------------------------------
// Kernel A: fused depthwise-conv(3) + residual + ReLU + GEMV(lin_w) + sigmoid
// One block = one (batch, 16-timestep) tile. 8 waves each own a 64-wide D
// slice; lin_w is replicated across all 16 WMMA columns so the B fragment is
// branch-free, and column N=0 of the f32 accumulator carries the logits.
// ---------------------------------------------------------------------------
__global__ __launch_bounds__(256) void cif_alpha_kernel(
    const float* __restrict__ hidden, const float* __restrict__ mask,
    const float* __restrict__ conv_w, const float* __restrict__ conv_b,
    const float* __restrict__ lin_w,  const float* __restrict__ lin_b,
    float* __restrict__ alphasM)
{
    __shared__ __align__(16) float sh[18 * SHSTRIDE]; // rows t0-1 .. t0+16
    __shared__ float swz[3 * D_];
    __shared__ float sb[D_];
    __shared__ float slw[D_];
    __shared__ float part[8][16];

    const int tile = blockIdx.x;
    const int b    = blockIdx.y;
    const int t0   = tile * 16;
    const int tid  = threadIdx.x;

    for (int i = tid; i < 3 * D_; i += 256) swz[i] = conv_w[i];
    for (int i = tid; i < D_; i += 256) { sb[i] = conv_b[i]; slw[i] = lin_w[i]; }

    // Stage 18x512 hidden tile as float4 (b128). In-range rows go through the
    // async global->LDS path (global_load_async_to_lds_b128); edge rows get
    // zero-filled with plain LDS stores.
    for (int i = tid; i < 18 * (D_ / 4); i += 256) {
        const int r  = i >> 7;          // row 0..17
        const int c4 = i & 127;         // float4 column
        const int t  = t0 - 1 + r;
        float* ldst = sh + r * SHSTRIDE + c4 * 4;
        if (t >= 0 && t < T_) {
            const float* gsrc = hidden + ((size_t)b * T_ + t) * D_ + c4 * 4;
#ifdef HAVE_ASYNC_LDS
            __builtin_amdgcn_global_load_async_to_lds_b128(
                (__attribute__((address_space(1))) v4i*)(void*)gsrc,
                (__attribute__((address_space(3))) v4i*)(void*)ldst, 0, 0);
#else
            *(float4*)ldst = *(const float4*)gsrc;
#endif
        } else {
            float4 z; z.x = 0.f; z.y = 0.f; z.z = 0.f; z.w = 0.f;
            *(float4*)ldst = z;
        }
    }
#ifdef HAVE_ASYNC_LDS
#if __has_builtin(__builtin_amdgcn_s_wait_asynccnt)
    __builtin_amdgcn_s_wait_asynccnt(0);
#else
    asm volatile("s_wait_asynccnt 0x0" ::: "memory");
#endif
#endif
    __syncthreads();

    const int wave   = tid >> 5;
    const int lane   = tid & 31;
    const int n      = lane & 15;   // A-matrix row M (and B/D column N)
    const int half   = lane >> 4;   // K half select per 16x4 f32 layout
    const int dstart = wave * 64;

    v8f c = {};
    for (int kc = 0; kc < 16; ++kc) {
        const int dbase = dstart + kc * 4;
        v2f a, bf;
        #pragma unroll
        for (int j = 0; j < 2; ++j) {
            const int d  = dbase + 2 * half + j;          // A lane: K = 2*half + j
            const float hm1 = sh[(n + 0) * SHSTRIDE + d]; // t-1
            const float h0  = sh[(n + 1) * SHSTRIDE + d]; // t
            const float hp1 = sh[(n + 2) * SHSTRIDE + d]; // t+1
            float mem = fmaf(swz[d * 3 + 0], hm1,
                        fmaf(swz[d * 3 + 1], h0,
                        fmaf(swz[d * 3 + 2], hp1, sb[d])));
            a[j] = fmaxf(mem + h0, 0.f);                  // relu(conv + residual)
        }
        // lin_w replicated across all 16 columns -> branch-free B fragment
        bf.x = slw[dbase + 0 + half];                     // K = half
        bf.y = slw[dbase + 2 + half];                     // K = 2 + half
        c = __builtin_amdgcn_wmma_f32_16x16x4_f32(
                false, a, false, bf, (short)0, c, false, false);
    }
    // Column N=0: lane 0 holds M=0..7 in c[0..7], lane 16 holds M=8..15
    if (lane == 0) {
        #pragma unroll
        for (int r = 0; r < 8; ++r) part[wave][r] = c[r];
    }
    if (lane == 16) {
        #pragma unroll
        for (int r = 0; r < 8; ++r) part[wave][8 + r] = c[r];
    }
    __syncthreads();

    if (tid < 16) {
        float s = 0.f;
        #pragma unroll
        for (int w = 0; w < 8; ++w) s += part[w][tid];
        const float logit = s + lin_b[0];
        float alpha = 1.f / (1.f + expf(-logit));
        alpha = fmaxf(alpha * 1.0f - 0.0f, 0.f);          // SMOOTH=1, NOISE=0
        alpha *= mask[(size_t)b * T_ + t0 + tid];
        alphasM[(size_t)b * T_ + t0 + tid] = alpha;
    }
}

// ---------------------------------------------------------------------------
// Kernel B: sequential CIF scan. One lane per batch (16 active lanes),
// factors every output frame into per-t coefficients + fire-position list.
// ---------------------------------------------------------------------------
__global__ __launch_bounds__(32) void cif_scan_kernel(
    const float* __restrict__ alphasM, const float* __restrict__ mask,
    float* __restrict__ o_alphas2, float* __restrict__ o_peak,
    float* __restrict__ o_tok,
    float* __restrict__ curw, float* __restrict__ leadw,
    int* __restrict__ fpos, int* __restrict__ nf)
{
    const int b = threadIdx.x;
    if (b >= B_) return;
    float integ = 0.f, asum = 0.f;
    int k = 0;
    for (int t = 0; t < T1_; ++t) {
        const float aM = (t < T_) ? alphasM[(size_t)b * T_ + t] : 0.f;
        const float m1 = (t < T_) ? mask[(size_t)b * T_ + t] : 0.f;
        const float m2 = (t == 0) ? 1.f : mask[(size_t)b * T_ + (t - 1)];
        const float alpha = aM + (m2 - m1) * TAIL_;        // tail term
        o_alphas2[b * T1_ + t] = alpha;
        asum += alpha;
        const float dist = 1.f - integ;
        integ += alpha;
        const float fires = integ;
        const bool fire = (integ >= 1.0f);
        if (fire) integ -= 1.f;
        const float cur = fire ? dist : alpha;
        const float rem = alpha - cur;
        o_peak[b * T1_ + t] = fires;
        curw[b * T1_ + t]   = cur;
        if (fire) { fpos[b * T1_ + k] = t; leadw[b * T1_ + k] = rem; ++k; }
    }
    nf[b] = k;
    o_tok[b] = floorf(asum);
}

// ---------------------------------------------------------------------------
// Kernel C: gather. Block (b,k) builds acoustic frame k of batch b as
// rem_{prev fire} * h[prev] + sum cur_t * h[t] over its segment; k >= #fires
// writes zeros (also zero-inits the poisoned output). float4 (b128) traffic.
// ---------------------------------------------------------------------------
__global__ __launch_bounds__(128) void cif_gather_kernel(
    const float* __restrict__ hidden, const float* __restrict__ curw,
    const float* __restrict__ leadw,  const int* __restrict__ fpos,
    const int* __restrict__ nf, float* __restrict__ o_ac)
{
    const int b   = blockIdx.y;
    const int k   = blockIdx.x;
    const int tid = threadIdx.x;          // 0..127, one float4 each
    float ax = 0.f, ay = 0.f, az = 0.f, aw = 0.f;
    const int nfb = nf[b];
    if (k < nfb) {
        const int tk    = fpos[b * T1_ + k];
        const int tprev = (k > 0) ? fpos[b * T1_ + k - 1] : -1;
        if (k > 0 && tprev < T_) {
            const float lr = leadw[b * T1_ + k - 1];
            const float4 v = ((const float4*)(hidden + ((size_t)b * T_ + tprev) * D_))[tid];
            ax = lr * v.x; ay = lr * v.y; az = lr * v.z; aw = lr * v.w;
        }
        for (int t = tprev + 1; t <= tk && t < T_; ++t) {
            const float cw = curw[b * T1_ + t];
            const float* hrow = hidden + ((size_t)b * T_ + t) * D_;
            __builtin_prefetch(hrow + D_ + tid * 4, 0, 1);  // global_prefetch_b8
            const float4 v = ((const float4*)hrow)[tid];
            ax = fmaf(cw, v.x, ax); ay = fmaf(cw, v.y, ay);
            az = fmaf(cw, v.z, az); aw = fmaf(cw, v.w, aw);
        }
    }
    float4 o; o.x = ax; o.y = ay; o.z = az; o.w = aw;
    ((float4*)(o_ac + ((size_t)b * T1_ + k) * D_))[tid] = o;
}

// ---------------------------------------------------------------------------
extern "C" void kernel_launch(void* const* d_in, const int* in_sizes, int n_in,
                              void* d_out, int out_size, void* d_ws, size_t ws_size,
                              hipStream_t stream)
{
    const float* hidden = (const float*)d_in[0];
    const float* mask   = (const float*)d_in[1];
    const float* conv_w = (const float*)d_in[2];
    const float* conv_b = (const float*)d_in[3];
    const float* lin_w  = (const float*)d_in[4];
    const float* lin_b  = (const float*)d_in[5];

    // d_out: acoustic_embeds (16*1025*512) | token_num (16) | alphas2 (16*1025) | cif_peak (16*1025)
    float* o_ac  = (float*)d_out;
    float* o_tok = o_ac + (size_t)B_ * T1_ * D_;
    float* o_al  = o_tok + B_;
    float* o_pk  = o_al + B_ * T1_;

    // workspace: alphasM (B*T) | curw (B*T1) | leadw (B*T1) | fpos (B*T1 int) | nf (B int)
    float* ws      = (float*)d_ws;
    float* alphasM = ws;
    float* curw    = alphasM + B_ * T_;
    float* leadw   = curw + B_ * T1_;
    int*   fpos    = (int*)(leadw + B_ * T1_);
    int*   nf      = fpos + B_ * T1_;

    cif_alpha_kernel<<<dim3(T_ / 16, B_), 256, 0, stream>>>(
        hidden, mask, conv_w, conv_b, lin_w, lin_b, alphasM);
    cif_scan_kernel<<<1, 32, 0, stream>>>(
        alphasM, mask, o_al, o_pk, o_tok, curw, leadw, fpos, nf);
    cif_gather_kernel<<<dim3(T1_, B_), 128, 0, stream>>>(
        hidden, curw, leadw, fpos, nf, o_ac);
}